// STGCN_model_35115652612671
// MI455X (gfx1250) — compile-verified
//
#include <hip/hip_runtime.h>

#define IN_F 128
#define HID 32

typedef __attribute__((ext_vector_type(2))) float v2f;
typedef __attribute__((ext_vector_type(8))) float v8f;

// ---------------- degree / normalization ----------------

__global__ void deg_init_kernel(float* __restrict__ deg, int n) {
    int i = blockIdx.x * blockDim.x + threadIdx.x;
    if (i < n) deg[i] = 1.0f;  // self-loop contributes weight 1
}

__global__ void deg_accum_kernel(const long long* __restrict__ ei,
                                 const float* __restrict__ w,
                                 float* __restrict__ deg, int n_edges) {
    int e = blockIdx.x * blockDim.x + threadIdx.x;
    if (e < n_edges) {
        int dst = (int)ei[(long long)n_edges + e];  // row = aggregation target
        atomicAdd(&deg[dst], w[e]);
    }
}

__global__ void dinv_kernel(float* __restrict__ deg, int n) {
    int i = blockIdx.x * blockDim.x + threadIdx.x;
    if (i < n) {
        float d = deg[i];
        deg[i] = (d > 0.0f) ? rsqrtf(d) : 0.0f;  // in-place -> dinv
    }
}

// ---------------- h = x @ W1 via fp32 WMMA ----------------
// One wave -> one 16x16 tile of h. K-loop: 32 x v_wmma_f32_16x16x4_f32.
// A (16x4 f32): lanes 0-15 hold M=lane, K={0,1}; lanes 16-31 hold M=lane-16, K={2,3}.
// B (4x16 f32): mirrored — lane half selects K pair, lane&15 selects N.
// C/D (16x16 f32): VGPR g, lanes 0-15 -> M=g, lanes 16-31 -> M=g+8; N=lane&15.

__global__ void __launch_bounds__(256)
gemm_xw_kernel(const float* __restrict__ x, const float* __restrict__ W1,
               float* __restrict__ h, int n_nodes) {
    int wave = (blockIdx.x * blockDim.x + threadIdx.x) >> 5;  // wave32
    int lane = threadIdx.x & 31;
    int ntiles = (n_nodes / 16) * (HID / 16);
    if (wave >= ntiles) return;  // wave-uniform: EXEC stays all-ones for WMMA

    int mt = wave >> 1;
    int nt = wave & 1;
    int m0 = mt * 16;
    int n0 = nt * 16;
    int half = lane >> 4;   // 0: K pair {0,1}; 1: K pair {2,3}
    int l15  = lane & 15;

    const float* xr = x + (size_t)(m0 + l15) * IN_F;  // A row for this lane
    const float* wc = W1 + n0 + l15;                  // B column for this lane

    v8f acc = {};
    for (int k = 0; k < IN_F; k += 4) {
        int kb = k + half * 2;
        v2f a, b;
        a.x = xr[kb];
        a.y = xr[kb + 1];
        b.x = wc[(size_t)kb * HID];
        b.y = wc[(size_t)(kb + 1) * HID];
        // (neg_a, A, neg_b, B, c_mod, C, reuse_a, reuse_b)
        acc = __builtin_amdgcn_wmma_f32_16x16x4_f32(
            false, a, false, b, (short)0, acc, false, false);
    }

    float* hr = h + (size_t)(m0 + half * 8) * HID + n0 + l15;
#pragma unroll
    for (int g = 0; g < 8; ++g)
        hr[(size_t)g * HID] = acc[g];
}

// ---------------- aggregation ----------------

// agg[n][c] = dinv[n]^2 * h[n][c]   (self-loop term; also zero-initializes agg)
__global__ void agg_init_kernel(const float* __restrict__ h,
                                const float* __restrict__ dinv,
                                float* __restrict__ agg, int n_nodes) {
    int t = blockIdx.x * blockDim.x + threadIdx.x;
    if (t < n_nodes * HID) {
        float di = dinv[t >> 5];
        agg[t] = di * di * h[t];
    }
}

// 32 lanes per edge (one per channel): coalesced gather of h[src], coalesced
// f32 atomic scatter into agg[dst]. Per-edge scalars are wave-uniform.
__global__ void edge_msg_kernel(const long long* __restrict__ ei,
                                const float* __restrict__ w,
                                const float* __restrict__ dinv,
                                const float* __restrict__ h,
                                float* __restrict__ agg, int n_edges) {
    long long t = (long long)blockIdx.x * blockDim.x + threadIdx.x;
    if (t < (long long)n_edges * HID) {
        int e  = (int)(t >> 5);
        int ch = (int)(t & 31);
        int src = (int)ei[e];                       // col = message source
        int dst = (int)ei[(long long)n_edges + e];  // row = target
        float norm = dinv[dst] * w[e] * dinv[src];
        atomicAdd(&agg[(size_t)dst * HID + ch], norm * h[(size_t)src * HID + ch]);
    }
}

// ---------------- out = relu(agg + b1) @ W2 + b2 ----------------
// One wave per node; 32 lanes = 32 channels; butterfly shfl_xor reduction.
__global__ void out_kernel(const float* __restrict__ agg,
                           const float* __restrict__ b1,
                           const float* __restrict__ W2,
                           const float* __restrict__ b2,
                           float* __restrict__ out, int n_nodes) {
    int t = blockIdx.x * blockDim.x + threadIdx.x;
    int n  = t >> 5;
    int ch = t & 31;
    if (n >= n_nodes) return;
    float v = agg[(size_t)n * HID + ch] + b1[ch];
    v = v > 0.0f ? v : 0.0f;
    v *= W2[ch];
#pragma unroll
    for (int off = 16; off > 0; off >>= 1)
        v += __shfl_xor(v, off, 32);
    if (ch == 0) out[n] = v + b2[0];
}

// ---------------- launcher ----------------

extern "C" void kernel_launch(void* const* d_in, const int* in_sizes, int n_in,
                              void* d_out, int out_size, void* d_ws, size_t ws_size,
                              hipStream_t stream) {
    const float*     x   = (const float*)d_in[0];
    const long long* ei  = (const long long*)d_in[1];  // int64 [2, E] flat
    const float*     w   = (const float*)d_in[2];
    const float*     W1  = (const float*)d_in[3];
    const float*     b1  = (const float*)d_in[4];
    const float*     W2  = (const float*)d_in[5];
    const float*     b2  = (const float*)d_in[6];
    float*           out = (float*)d_out;

    int n_nodes = in_sizes[0] / IN_F;
    int n_edges = in_sizes[2];

    // workspace: deg/dinv [N] | h [N*32] | agg [N*32]  (~26 MB)
    float* deg = (float*)d_ws;
    float* h   = deg + n_nodes;
    float* agg = h + (size_t)n_nodes * HID;

    const int B = 256;
    deg_init_kernel<<<(n_nodes + B - 1) / B, B, 0, stream>>>(deg, n_nodes);
    deg_accum_kernel<<<(n_edges + B - 1) / B, B, 0, stream>>>(ei, w, deg, n_edges);
    dinv_kernel<<<(n_nodes + B - 1) / B, B, 0, stream>>>(deg, n_nodes);

    int ntiles = (n_nodes / 16) * (HID / 16);
    gemm_xw_kernel<<<(ntiles * 32 + B - 1) / B, B, 0, stream>>>(x, W1, h, n_nodes);

    agg_init_kernel<<<((size_t)n_nodes * HID + B - 1) / B, B, 0, stream>>>(h, deg, agg, n_nodes);

    long long etot = (long long)n_edges * HID;
    edge_msg_kernel<<<(unsigned)((etot + B - 1) / B), B, 0, stream>>>(ei, w, deg, h, agg, n_edges);

    out_kernel<<<((size_t)n_nodes * 32 + B - 1) / B, B, 0, stream>>>(agg, b1, W2, b2, out, n_nodes);
}